// REDecoder_89635967468130
// MI455X (gfx1250) — compile-verified
//
#include <hip/hip_runtime.h>
#include <math.h>

// Problem constants from the reference
#define BB 8
#define TT 512
#define DD 256
#define SS 64
#define PP 1024
#define RR 12
#define NN (RR * DD)          // 3072 GEMM N dimension
#define NROWS (BB * SS)       // 512 pooled rows

typedef __attribute__((ext_vector_type(2))) float v2f;
typedef __attribute__((ext_vector_type(8))) float v8f;

#define LSTR 260   // padded LDS row stride (floats): 260 % 64 == 4 -> conflict-free A reads

// ---------------------------------------------------------------------------
// Kernel 1: ragged span max-pooling + per-span linear terms.
// One block per (b,s): 256 threads (8 waves); thread d owns channel d.
// lin[bs*24 + r]      = pooled[bs,:]·W_linear[0:D, r]      (head part)
// lin[bs*24 + 12 + r] = pooled[bs,:]·W_linear[D:2D, r]     (tail part)
// ---------------------------------------------------------------------------
__global__ __launch_bounds__(256) void span_pool_kernel(
    const float* __restrict__ encoded,   // [B,T,D]
    const int*   __restrict__ starts,    // [B,S]
    const int*   __restrict__ lens,      // [B,S]
    const float* __restrict__ W_linear,  // [2D,R]
    float*       __restrict__ pooled,    // [B,S,D]
    float*       __restrict__ lin)       // [B,S,24]
{
    __shared__ float rowLDS[DD];

    const int bs = blockIdx.x;           // b*S + s
    const int d  = threadIdx.x;          // 0..255
    const int b  = bs >> 6;
    const int st = starts[bs];
    const int en = st + lens[bs] + 1;    // exclusive, always >= st+1
    const float* base = encoded + ((size_t)b * TT) * DD + d;
    float m = -3.402823466e+38f;
    for (int t = st; t < en; ++t)
        m = fmaxf(m, base[(size_t)t * DD]);
    pooled[(size_t)bs * DD + d] = m;
    rowLDS[d] = m;
    __syncthreads();

    // 8 waves x 3 outputs each = 24 linear dot products of length 256
    const int wid  = threadIdx.x >> 5;
    const int lane = threadIdx.x & 31;
    #pragma unroll
    for (int oo = 0; oo < 3; ++oo) {
        const int o   = wid * 3 + oo;          // 0..23
        const int r   = (o < 12) ? o : (o - 12);
        const int off = (o < 12) ? 0 : DD;
        float pl = 0.f;
        #pragma unroll
        for (int j = 0; j < 8; ++j) {
            const int dd = lane + 32 * j;
            pl += rowLDS[dd] * W_linear[(off + dd) * RR + r];
        }
        pl += __shfl_xor(pl, 16, 32);
        pl += __shfl_xor(pl,  8, 32);
        pl += __shfl_xor(pl,  4, 32);
        pl += __shfl_xor(pl,  2, 32);
        pl += __shfl_xor(pl,  1, 32);
        if (lane == 0) lin[bs * 24 + o] = pl;
    }
}

// ---------------------------------------------------------------------------
// Kernel 2: G[512, 3072] = pooled[512, 256] @ W_cat[256, 3072] via
// wmma f32 16x16x4.  W_cat[d, r*256+e] = W_bilinear[r,d,e].
// grid = (32 M-tiles, 12 N-chunks), block = 128 (4 waves x 4 N-tiles).
// ---------------------------------------------------------------------------
__global__ __launch_bounds__(128) void span_gemm_kernel(
    const float* __restrict__ pooled,     // [512, 256]
    const float* __restrict__ W_bilinear, // [R,D,D]
    float*       __restrict__ G)          // [512, 3072]
{
    __shared__ float aLDS[16 * LSTR];

    const int bs0 = blockIdx.x * 16;      // first pooled row of this M-tile
    const int tid = threadIdx.x;

    // ---- stage A tile (16 x 256) into LDS, float4, padded stride ----
    for (int idx = tid; idx < 16 * 64; idx += 128) {
        const int row = idx >> 6;
        const int c4  = (idx & 63) << 2;
        const float4 av = *(const float4*)&pooled[((size_t)(bs0 + row)) * DD + c4];
        *(float4*)&aLDS[row * LSTR + c4] = av;   // 1040*row bytes: 16B aligned
    }
    __syncthreads();

    const int wid  = tid >> 5;
    const int lane = tid & 31;
    const int l16  = lane & 15;
    const int hi   = lane >> 4;

    for (int sub = 0; sub < 4; ++sub) {
        const int ntile = blockIdx.y * 16 + wid * 4 + sub;   // 0..191
        const int n0    = ntile * 16;
        const int n     = n0 + l16;                          // global N column
        // W_cat column n lives at W_bilinear[(n>>8)*65536 + (n&255)], K-stride 256
        const float* wp = W_bilinear + (((size_t)(n >> 8)) << 16) + (n & 255)
                                     + (hi ? 2 * DD : 0);
        v8f c = {};
        #pragma unroll 4
        for (int k = 0; k < 64; ++k) {
            // A frag: lanes 0-15 -> K = 4k,4k+1 ; lanes 16-31 -> K = 4k+2,4k+3
            const float* ap = &aLDS[l16 * LSTR + 4 * k + 2 * hi];
            v2f a;  a.x = ap[0]; a.y = ap[1];                 // 8B-aligned ds b64
            v2f bf; bf.x = wp[(size_t)k * 1024];              // row 4k
                    bf.y = wp[(size_t)k * 1024 + DD];         // row 4k+1
            c = __builtin_amdgcn_wmma_f32_16x16x4_f32(
                    false, a, false, bf, (short)0, c, false, false);
        }
        // D layout: lane half hi -> rows v + 8*hi, column n0 + l16
        #pragma unroll
        for (int v = 0; v < 8; ++v) {
            const int m = v + 8 * hi;
            G[((size_t)(bs0 + m)) * NN + n0 + l16] = c[v];
        }
    }
}

// ---------------------------------------------------------------------------
// Kernel 3: per-pair scoring.  One wave per pair (4 pairs / 128-thread block).
// score[bp,r] = sigmoid( b[r] + linH[head,r] + linT[tail,r]
//                        + sum_e G[head, r*256+e] * pooled[tail, e] )
// ---------------------------------------------------------------------------
__global__ __launch_bounds__(128) void pair_score_kernel(
    const float* __restrict__ pooled,    // [512, 256]
    const float* __restrict__ G,         // [512, 3072]
    const float* __restrict__ lin,       // [512, 24]
    const float* __restrict__ b_linear,  // [R]
    const int*   __restrict__ pair_head, // [B,P]
    const int*   __restrict__ pair_tail, // [B,P]
    float*       __restrict__ out)       // [B,P,R]
{
    const int wid  = threadIdx.x >> 5;
    const int lane = threadIdx.x & 31;
    const int bp   = blockIdx.x * 4 + wid;       // 0..8191
    const int b    = bp >> 10;
    const int h    = pair_head[bp];
    const int t    = pair_tail[bp];
    const int hrow = b * SS + h;
    const int trow = b * SS + t;

    // lane owns e = lane*8 .. lane*8+7 (contiguous -> float4 pairs)
    const float* tp = pooled + ((size_t)trow) * DD + lane * 8;
    const float4 t0 = *(const float4*)tp;
    const float4 t1 = *(const float4*)(tp + 4);

    const float* gr = G + ((size_t)hrow) * NN + lane * 8;
    float acc[RR];
    #pragma unroll
    for (int r = 0; r < RR; ++r) {
        const float4 g0 = *(const float4*)(gr + r * DD);
        const float4 g1 = *(const float4*)(gr + r * DD + 4);
        float s = g0.x * t0.x + g0.y * t0.y + g0.z * t0.z + g0.w * t0.w
                + g1.x * t1.x + g1.y * t1.y + g1.z * t1.z + g1.w * t1.w;
        acc[r] = s;
    }
    // reduce each of the 12 partials across the wave
    #pragma unroll
    for (int r = 0; r < RR; ++r) {
        float s = acc[r];
        s += __shfl_xor(s, 16, 32);
        s += __shfl_xor(s,  8, 32);
        s += __shfl_xor(s,  4, 32);
        s += __shfl_xor(s,  2, 32);
        s += __shfl_xor(s,  1, 32);
        acc[r] = s;
    }
    if (lane < RR) {
        const float sc = acc[lane] + b_linear[lane]
                       + lin[hrow * 24 + lane]          // head · Wl[0:D, r]
                       + lin[trow * 24 + 12 + lane];    // tail · Wl[D:2D, r]
        out[(size_t)bp * RR + lane] = 1.f / (1.f + __expf(-sc));
    }
}

// ---------------------------------------------------------------------------
extern "C" void kernel_launch(void* const* d_in, const int* in_sizes, int n_in,
                              void* d_out, int out_size, void* d_ws, size_t ws_size,
                              hipStream_t stream) {
    const float* encoded    = (const float*)d_in[0];  // [B,T,D]
    const float* W_linear   = (const float*)d_in[1];  // [2D,R]
    const float* b_linear   = (const float*)d_in[2];  // [R]
    const float* W_bilinear = (const float*)d_in[3];  // [R,D,D]
    const int*   span_starts= (const int*)d_in[4];    // [B,S]
    const int*   span_lens  = (const int*)d_in[5];    // [B,S]
    const int*   pair_head  = (const int*)d_in[6];    // [B,P]
    const int*   pair_tail  = (const int*)d_in[7];    // [B,P]
    float*       out        = (float*)d_out;          // [B,P,R]

    char* ws = (char*)d_ws;
    float* pooled = (float*)ws;                                   // 512*256*4   = 512 KB
    float* lin    = (float*)(ws + (size_t)NROWS * DD * 4);        // 512*24*4    =  48 KB
    float* G      = (float*)(ws + (size_t)NROWS * (DD + 24) * 4); // 512*3072*4  =   6 MB

    span_pool_kernel<<<NROWS, DD, 0, stream>>>(encoded, span_starts, span_lens,
                                               W_linear, pooled, lin);

    dim3 ggrid(NROWS / 16, NN / 256);   // (32, 12)
    span_gemm_kernel<<<ggrid, 128, 0, stream>>>(pooled, W_bilinear, G);

    pair_score_kernel<<<(BB * PP) / 4, 128, 0, stream>>>(pooled, G, lin, b_linear,
                                                         pair_head, pair_tail, out);
}